// Agent_14903536517121
// MI455X (gfx1250) — compile-verified
//
#include <hip/hip_runtime.h>
#include <hip/hip_bf16.h>
#include <math.h>

typedef __attribute__((ext_vector_type(2))) float v2f;
typedef __attribute__((ext_vector_type(4))) float v4f;
typedef __attribute__((ext_vector_type(8))) float v8f;

#define B_BATCH 2048
#define FEATD   512
#define NRING   2
#define NNEUR   2048
#define NACT    7
#define NUPDATE 10
#define ALPHA   0.2f          // DT/TAU
#define RS_STRIDE 2052        // 2048 + 4 pad -> stride % 64 == 4, conflict-free A reads
#define MROWS   32            // batch rows per block (two 16-row WMMA M-tiles)

// ---------------------------------------------------------------------------
// 1) CNN encoder: per-image scalar kernel (tiny: 7x7x3 input). Output h (B,1024)
//    in torch-flatten (C,H,W) order.
// ---------------------------------------------------------------------------
__global__ void __launch_bounds__(128) conv_encoder_kernel(
    const float* __restrict__ x,
    const float* __restrict__ w1, const float* __restrict__ b1,
    const float* __restrict__ w2, const float* __restrict__ b2,
    const float* __restrict__ w3, const float* __restrict__ b3,
    float* __restrict__ h_out)
{
  __shared__ float o1[16 * 36];  // (16,6,6)
  __shared__ float o2[32 * 25];  // (32,5,5)
  const int b   = blockIdx.x;
  const int tid = threadIdx.x;

  // conv1: (B,7,7,3) NHWC input, OIHW weights (16,3,2,2) -> (16,6,6)
  for (int idx = tid; idx < 16 * 36; idx += 128) {
    const int o = idx / 36, rem = idx % 36, i = rem / 6, j = rem % 6;
    float acc = b1[o];
    #pragma unroll
    for (int c = 0; c < 3; ++c)
      #pragma unroll
      for (int kh = 0; kh < 2; ++kh)
        #pragma unroll
        for (int kw = 0; kw < 2; ++kw)
          acc += x[((b * 7 + (i + kh)) * 7 + (j + kw)) * 3 + c] *
                 w1[((o * 3 + c) * 2 + kh) * 2 + kw];
    o1[idx] = fmaxf(acc, 0.0f);
  }
  __syncthreads();

  // conv2: (16,6,6) -> (32,5,5), weights (32,16,2,2)
  for (int idx = tid; idx < 32 * 25; idx += 128) {
    const int o = idx / 25, rem = idx % 25, i = rem / 5, j = rem % 5;
    float acc = b2[o];
    for (int c = 0; c < 16; ++c)
      #pragma unroll
      for (int kh = 0; kh < 2; ++kh)
        #pragma unroll
        for (int kw = 0; kw < 2; ++kw)
          acc += o1[c * 36 + (i + kh) * 6 + (j + kw)] *
                 w2[((o * 16 + c) * 2 + kh) * 2 + kw];
    o2[idx] = fmaxf(acc, 0.0f);
  }
  __syncthreads();

  // conv3: (32,5,5) -> (64,4,4) = 1024, weights (64,32,2,2)
  for (int idx = tid; idx < 64 * 16; idx += 128) {
    const int o = idx / 16, rem = idx % 16, i = rem / 4, j = rem % 4;
    float acc = b3[o];
    for (int c = 0; c < 32; ++c)
      #pragma unroll
      for (int kh = 0; kh < 2; ++kh)
        #pragma unroll
        for (int kw = 0; kw < 2; ++kw)
          acc += o2[c * 25 + (i + kh) * 5 + (j + kw)] *
                 w3[((o * 32 + c) * 2 + kh) * 2 + kw];
    h_out[(size_t)b * 1024 + idx] = fmaxf(acc, 0.0f);  // idx == c*16+i*4+j
  }
}

// ---------------------------------------------------------------------------
// 2) Weight transpose: in (R,C) row-major -> out (C,R) row-major
// ---------------------------------------------------------------------------
__global__ void transpose_kernel(const float* __restrict__ in, float* __restrict__ out,
                                 int R, int C)
{
  int idx = blockIdx.x * blockDim.x + threadIdx.x;
  if (idx < R * C) {
    int r = idx / C, c = idx % C;
    out[(size_t)c * R + r] = in[idx];
  }
}

// ---------------------------------------------------------------------------
// 3) Generic fp32 WMMA GEMM: C[M,N] = act(A[M,K] @ Bm[K,N] + bias)
//    4 waves/block, each wave owns a 16x16 tile; grid (M/16, N/64).
// ---------------------------------------------------------------------------
__global__ void __launch_bounds__(128, 1) wmma_gemm_kernel(
    const float* __restrict__ A, const float* __restrict__ Bm,
    const float* __restrict__ bias, float* __restrict__ C,
    int M, int N, int K, int do_relu)
{
  const int wave = threadIdx.x >> 5;
  const int lane = threadIdx.x & 31;
  const int m0   = blockIdx.x * 16;
  const int n0   = blockIdx.y * 64 + wave * 16;
  const int mrow = lane & 15;           // A: lanes 0-15 M=0..15, 16-31 M=0..15
  const int kk   = (lane >> 4) << 1;    // A/B: upper half-wave holds K+2,K+3
  const int ncol = n0 + (lane & 15);

  const float* __restrict__ arow = A + (size_t)(m0 + mrow) * K + kk;

  v8f acc = (v8f)0.0f;
  for (int k = 0; k < K; k += 4) {
    // A: K,K+1 contiguous per lane -> single global_load_b64
    v2f a = *reinterpret_cast<const v2f*>(arow + k);
    v2f bf;
    bf.x = Bm[(size_t)(k + kk) * N + ncol];
    bf.y = Bm[(size_t)(k + kk + 1) * N + ncol];
    acc = __builtin_amdgcn_wmma_f32_16x16x4_f32(false, a, false, bf,
                                                (short)0, acc, false, false);
  }
  const float bv  = bias ? bias[ncol] : 0.0f;
  const int   mhi = (lane >> 4) << 3;   // C/D: VGPR v -> M = v (+8 upper half)
  #pragma unroll
  for (int v = 0; v < 8; ++v) {
    float r = acc[v] + bv;
    if (do_relu) r = fmaxf(r, 0.0f);
    C[(size_t)(m0 + v + mhi) * N + ncol] = r;
  }
}

// ---------------------------------------------------------------------------
// 4) Fused ring-attractor recurrence. One block = (ring r, 32 batch rows).
//    rs tile (32 x 2048 f32 ~ 256.5 KB) lives in LDS across all 10 Euler steps;
//    W_rec streamed from L2 (33.5 MB << 192 MB). Each wave owns 256 output
//    columns and BOTH 16-row M-tiles: every B fragment feeds 2 wmmas
//    (1 VMEM dword per wmma, ~16 flop per L2 byte).
// ---------------------------------------------------------------------------
__global__ void __launch_bounds__(256, 1) ring_recurrence_kernel(
    const float* __restrict__ rs_in, const float* __restrict__ W_rec,
    const float* __restrict__ xp, float* __restrict__ rs_out)
{
  extern __shared__ float rs[];                 // [MROWS][RS_STRIDE]
  const int r  = blockIdx.x >> 6;               // ring (64 batch tiles per ring)
  const int b0 = (blockIdx.x & 63) * MROWS;     // batch tile start
  const int tid = threadIdx.x;

  // vectorized tile load: 32 x 2048 floats as float4 (both sides 16B-aligned)
  for (int idx = tid; idx < MROWS * (NNEUR / 4); idx += 256) {
    const int row = idx / (NNEUR / 4), c4 = idx - row * (NNEUR / 4);
    *reinterpret_cast<v4f*>(&rs[row * RS_STRIDE + c4 * 4]) =
        *reinterpret_cast<const v4f*>(
            &rs_in[((size_t)(b0 + row) * NRING + r) * NNEUR + c4 * 4]);
  }
  __syncthreads();

  const int lane   = tid & 31;
  const int wave   = tid >> 5;
  const int mrow   = lane & 15;
  const int kk     = (lane >> 4) << 1;
  const int n_base = wave * 256;
  const int ncol_l = lane & 15;
  const int mhi    = (lane >> 4) << 3;
  const float* __restrict__ Wr   = W_rec + (size_t)r * NNEUR * NNEUR;
  const float* __restrict__ Wcol = Wr + n_base + ncol_l;   // lane-fixed column base

  for (int iter = 0; iter < NUPDATE; ++iter) {
    v8f acc_lo[16], acc_hi[16];
    #pragma unroll
    for (int t = 0; t < 16; ++t) { acc_lo[t] = (v8f)0.0f; acc_hi[t] = (v8f)0.0f; }

    for (int k = 0; k < NNEUR; k += 4) {
      // A fragments for both M-halves: single ds_load_b64 each
      v2f a_lo = *reinterpret_cast<const v2f*>(&rs[mrow * RS_STRIDE + k + kk]);
      v2f a_hi = *reinterpret_cast<const v2f*>(&rs[(mrow + 16) * RS_STRIDE + k + kk]);
      const float* w0 = Wcol + (size_t)(k + kk) * NNEUR;
      const float* w1 = w0 + NNEUR;
      if ((iter == 0) && (k + 32 < NNEUR)) {
        // warm L2->L0 stream a few rows ahead (uniform branch: EXEC untouched)
        __builtin_prefetch(w0 + (size_t)8 * NNEUR, 0, 1);
      }
      #pragma unroll
      for (int t = 0; t < 16; ++t) {
        v2f bf;
        bf.x = w0[t * 16];
        bf.y = w1[t * 16];
        acc_lo[t] = __builtin_amdgcn_wmma_f32_16x16x4_f32(false, a_lo, false, bf,
                                                          (short)0, acc_lo[t], false, false);
        acc_hi[t] = __builtin_amdgcn_wmma_f32_16x16x4_f32(false, a_hi, false, bf,
                                                          (short)0, acc_hi[t], false, false);
      }
    }
    __syncthreads();  // all A-fragment reads done before rs is overwritten

    #pragma unroll
    for (int t = 0; t < 16; ++t) {
      const int n = n_base + t * 16 + ncol_l;
      #pragma unroll
      for (int v = 0; v < 8; ++v) {
        const int mr = v + mhi;
        {
          const float old = rs[mr * RS_STRIDE + n];
          float drive = acc_lo[t][v] + xp[(size_t)(b0 + mr) * NNEUR + n];
          drive = fmaxf(drive, 0.0f);
          rs[mr * RS_STRIDE + n] = (1.0f - ALPHA) * old + ALPHA * drive;
        }
        {
          const int mr2 = mr + 16;
          const float old = rs[mr2 * RS_STRIDE + n];
          float drive = acc_hi[t][v] + xp[(size_t)(b0 + mr2) * NNEUR + n];
          drive = fmaxf(drive, 0.0f);
          rs[mr2 * RS_STRIDE + n] = (1.0f - ALPHA) * old + ALPHA * drive;
        }
      }
    }
    __syncthreads();
  }

  for (int idx = tid; idx < MROWS * (NNEUR / 4); idx += 256) {
    const int row = idx / (NNEUR / 4), c4 = idx - row * (NNEUR / 4);
    *reinterpret_cast<v4f*>(
        &rs_out[((size_t)(b0 + row) * NRING + r) * NNEUR + c4 * 4]) =
        *reinterpret_cast<const v4f*>(&rs[row * RS_STRIDE + c4 * 4]);
  }
}

// ---------------------------------------------------------------------------
// 5) Readout: logits = mean_r(rs) @ proj_out^T, Gumbel-max sample (deterministic
//    hash noise), log-prob of action, entropy. One block per batch element.
// ---------------------------------------------------------------------------
__device__ __forceinline__ float hash_u01(unsigned s)
{
  s ^= s >> 17; s *= 0xed5ad4bbu;
  s ^= s >> 11; s *= 0xac4c1b51u;
  s ^= s >> 15; s *= 0x31848babu;
  s ^= s >> 14;
  float u = (float)(s >> 8) * (1.0f / 16777216.0f);
  return fminf(fmaxf(u, 1e-6f), 1.0f - 1e-6f);
}

__global__ void __launch_bounds__(256) readout_kernel(
    const float* __restrict__ rs_new, const float* __restrict__ pw,
    float* __restrict__ act_out, float* __restrict__ logp_out,
    float* __restrict__ ent_out)
{
  __shared__ float red[NACT * 256];
  const int b = blockIdx.x, tid = threadIdx.x;
  float pacc[NACT];
  #pragma unroll
  for (int a = 0; a < NACT; ++a) pacc[a] = 0.0f;

  const float* r0 = rs_new + (size_t)b * NRING * NNEUR;
  const float* r1 = r0 + NNEUR;
  for (int n = tid; n < NNEUR; n += 256) {
    const float v = 0.5f * (r0[n] + r1[n]);
    #pragma unroll
    for (int a = 0; a < NACT; ++a) pacc[a] += v * pw[a * NNEUR + n];
  }
  #pragma unroll
  for (int a = 0; a < NACT; ++a) red[a * 256 + tid] = pacc[a];
  __syncthreads();
  for (int s = 128; s > 0; s >>= 1) {
    if (tid < s)
      for (int a = 0; a < NACT; ++a)
        red[a * 256 + tid] += red[a * 256 + tid + s];
    __syncthreads();
  }
  if (tid == 0) {
    float logits[NACT];
    #pragma unroll
    for (int a = 0; a < NACT; ++a) logits[a] = red[a * 256];

    int best = 0; float bestv = -1e30f;
    #pragma unroll
    for (int a = 0; a < NACT; ++a) {
      const float u = hash_u01(0x2a000000u ^ (unsigned)(b * NACT + a));
      const float g = -logf(-logf(u));
      const float sv = logits[a] + g;
      if (sv > bestv) { bestv = sv; best = a; }
    }
    float m = logits[0];
    #pragma unroll
    for (int a = 1; a < NACT; ++a) m = fmaxf(m, logits[a]);
    float se = 0.0f;
    #pragma unroll
    for (int a = 0; a < NACT; ++a) se += expf(logits[a] - m);
    const float lse = m + logf(se);
    float ent = 0.0f;
    #pragma unroll
    for (int a = 0; a < NACT; ++a) {
      const float lp = logits[a] - lse;
      ent -= expf(lp) * lp;
    }
    act_out[b]  = (float)best;
    logp_out[b] = logits[best] - lse;
    ent_out[b]  = ent;
  }
}

// ---------------------------------------------------------------------------
extern "C" void kernel_launch(void* const* d_in, const int* in_sizes, int n_in,
                              void* d_out, int out_size, void* d_ws, size_t ws_size,
                              hipStream_t stream)
{
  const float* x         = (const float*)d_in[0];
  const float* rs_cur    = (const float*)d_in[1];
  const float* conv1_w   = (const float*)d_in[2];
  const float* conv1_b   = (const float*)d_in[3];
  const float* conv2_w   = (const float*)d_in[4];
  const float* conv2_b   = (const float*)d_in[5];
  const float* conv3_w   = (const float*)d_in[6];
  const float* conv3_b   = (const float*)d_in[7];
  const float* lin_w     = (const float*)d_in[8];
  const float* lin_b     = (const float*)d_in[9];
  const float* proj_in_w = (const float*)d_in[10];
  const float* W_rec     = (const float*)d_in[11];
  const float* proj_out_w= (const float*)d_in[12];

  // outputs (tuple concatenated flat): rs_new, action, log_prob, entropy
  float* out    = (float*)d_out;
  float* rs_out = out;
  float* actp   = rs_out + (size_t)B_BATCH * NRING * NNEUR;
  float* logpp  = actp + B_BATCH;
  float* entp   = logpp + B_BATCH;

  // workspace
  float* ws    = (float*)d_ws;
  float* h     = ws;                                   // B x 1024
  float* feats = h     + (size_t)B_BATCH * 1024;       // B x 512
  float* xp    = feats + (size_t)B_BATCH * FEATD;      // B x 2048
  float* linT  = xp    + (size_t)B_BATCH * NNEUR;      // 1024 x 512
  float* projT = linT  + (size_t)1024 * FEATD;         // 512 x 2048

  // 1) conv encoder
  conv_encoder_kernel<<<B_BATCH, 128, 0, stream>>>(
      x, conv1_w, conv1_b, conv2_w, conv2_b, conv3_w, conv3_b, h);

  // 2) transpose weights into K-major layout for WMMA B fragments
  {
    int n1 = FEATD * 1024;
    transpose_kernel<<<(n1 + 255) / 256, 256, 0, stream>>>(lin_w, linT, FEATD, 1024);
    int n2 = NNEUR * FEATD;
    transpose_kernel<<<(n2 + 255) / 256, 256, 0, stream>>>(proj_in_w, projT, NNEUR, FEATD);
  }

  // 3) feats = relu(h @ lin_w^T + lin_b)   : (2048x1024)@(1024x512)
  wmma_gemm_kernel<<<dim3(B_BATCH / 16, FEATD / 64), 128, 0, stream>>>(
      h, linT, lin_b, feats, B_BATCH, FEATD, 1024, 1);

  // 4) xp = feats @ proj_in_w^T            : (2048x512)@(512x2048)
  wmma_gemm_kernel<<<dim3(B_BATCH / 16, NNEUR / 64), 128, 0, stream>>>(
      feats, projT, nullptr, xp, B_BATCH, NNEUR, FEATD, 0);

  // 5) fused 10-step ring recurrence (dominant: ~343 GFLOP fp32 via v_wmma)
  {
    const size_t lds_bytes = (size_t)MROWS * RS_STRIDE * sizeof(float); // ~256.5 KB
    (void)hipFuncSetAttribute((const void*)ring_recurrence_kernel,
                              hipFuncAttributeMaxDynamicSharedMemorySize,
                              (int)lds_bytes);
    ring_recurrence_kernel<<<NRING * (B_BATCH / MROWS), 256, lds_bytes, stream>>>(
        rs_cur, W_rec, xp, rs_out);
  }

  // 6) readout: logits, gumbel-max action, log_prob, entropy
  readout_kernel<<<B_BATCH, 256, 0, stream>>>(rs_out, proj_out_w, actp, logpp, entp);
}